// AttentionAggregation_7224134992394
// MI455X (gfx1250) — compile-verified
//
#include <hip/hip_runtime.h>
#include <hip/hip_bf16.h>

// Problem constants (from reference): B=16, S=4096, H=1024, f32 everywhere.
#define BATCH 16
#define S_LEN 4096
#define H_DIM 1024
#define CHUNKS 16                 // j-chunks per batch in pass 3
#define CHUNK_J (S_LEN / CHUNKS)  // 256

typedef __attribute__((ext_vector_type(2))) float v2f;
typedef __attribute__((ext_vector_type(8))) float v8f;

// ---------------------------------------------------------------------------
// K1: scores[b][j] = am[b] * dot(hs[b, start+j, :], w)   for j < qlen
//
// One wave handles 16 rows. WMMA f32 16x16x4:
//   A (16x4)  = 16 hidden-state rows, K-chunk of 4. Lane L holds row (L&15),
//               K = {2t, 2t+1} with t = L>>4  -> one b64 load per lane.
//   B (4x16)  = w[k0..k0+3] replicated across all 16 columns. Per the C-matrix
//               analogous layout (VGPR r holds K=r lanes 0-15, K=r+2 lanes
//               16-31), lane L needs {w[k0+2t], w[k0+2t+1]} -> one b64 load.
//   C (16x16) = every column n holds the same 16 dot products.
// Two accumulators break the WMMA->WMMA dependency chain; merged elementwise.
// ---------------------------------------------------------------------------
__global__ __launch_bounds__(128) void aa_k1_scores(
    const float* __restrict__ hs, const int* __restrict__ qi,
    const float* __restrict__ am, const float* __restrict__ w,
    float* __restrict__ scores)
{
    const int b    = blockIdx.x;
    const int wv   = threadIdx.x >> 5;          // wave in block: 0..3
    const int lane = threadIdx.x & 31;
    const int j0   = (blockIdx.y * 4 + wv) * 16;

    const int start = qi[2 * b];
    const int qlen  = qi[2 * b + 1] - start;
    if (j0 >= qlen) return;                     // wave-uniform early exit

    const int m = lane & 15;                    // row within tile
    const int t = lane >> 4;                    // K-half select
    int row = j0 + m;
    row = (row < qlen) ? row : (qlen - 1);      // clamp: keep EXEC all-1s

    const float* rowp = hs + ((size_t)b * S_LEN + (size_t)(start + row)) * H_DIM
                           + 2 * t;
    const float* wp   = w + 2 * t;

    v8f c0 = {};
    v8f c1 = {};
    #pragma unroll 4
    for (int k0 = 0; k0 < H_DIM; k0 += 8) {
        v2f a0 = *(const v2f*)(rowp + k0);
        v2f b0 = *(const v2f*)(wp   + k0);
        c0 = __builtin_amdgcn_wmma_f32_16x16x4_f32(
                false, a0, false, b0, (short)0, c0, false, false);
        v2f a1 = *(const v2f*)(rowp + k0 + 4);
        v2f b1 = *(const v2f*)(wp   + k0 + 4);
        c1 = __builtin_amdgcn_wmma_f32_16x16x4_f32(
                false, a1, false, b1, (short)0, c1, false, false);
    }

    const float scale = am[b];
    // C layout: VGPR r = row r (lanes 0-15) / row r+8 (lanes 16-31), col = lane&15.
    // Column 0 lives in lanes 0 and 16 -> those lanes write 8 scores each.
    if (m == 0) {
        float* sp = scores + (size_t)b * S_LEN + j0 + t * 8;
        #pragma unroll
        for (int r = 0; r < 8; ++r) {
            float v0 = (c0[r] + c1[r]) * scale;
            sp[r] = v0;
        }
    }
}

// ---------------------------------------------------------------------------
// K2: per-batch softmax stats; rewrite scores in place as probabilities.
//   max_len = max_b(qlen_b) + 1 > qlen_b, so (max_len - qlen) zero-score
//   entries always exist: m = max(0, max_j s_j) and they add exp(-m) each to Z.
// ---------------------------------------------------------------------------
__global__ __launch_bounds__(256) void aa_k2_softmax(
    const int* __restrict__ qi, float* __restrict__ scores)
{
    __shared__ float red[256];
    const int b   = blockIdx.x;
    const int tid = threadIdx.x;

    const int start = qi[2 * b];
    const int qlen  = qi[2 * b + 1] - start;
    int max_len = 0;
    #pragma unroll
    for (int i = 0; i < BATCH; ++i) {
        int l = qi[2 * i + 1] + 1 - qi[2 * i];
        max_len = (l > max_len) ? l : max_len;
    }

    float* s = scores + (size_t)b * S_LEN;

    // pass 1: max
    float lm = -INFINITY;
    for (int j = tid; j < qlen; j += 256) lm = fmaxf(lm, s[j]);
    red[tid] = lm;
    __syncthreads();
    for (int off = 128; off > 0; off >>= 1) {
        if (tid < off) red[tid] = fmaxf(red[tid], red[tid + off]);
        __syncthreads();
    }
    const float mval = fmaxf(red[0], 0.0f);   // fold the implicit zero entries
    __syncthreads();

    // pass 2: sum of exp
    float ls = 0.0f;
    for (int j = tid; j < qlen; j += 256) ls += __expf(s[j] - mval);
    red[tid] = ls;
    __syncthreads();
    for (int off = 128; off > 0; off >>= 1) {
        if (tid < off) red[tid] += red[tid + off];
        __syncthreads();
    }
    const float Z    = red[0] + (float)(max_len - qlen) * __expf(-mval);
    const float invZ = 1.0f / Z;
    __syncthreads();

    // pass 3: probabilities in place
    for (int j = tid; j < qlen; j += 256) s[j] = __expf(s[j] - mval) * invZ;
}

// ---------------------------------------------------------------------------
// K3: partial[b][c][h] = sum_{j in chunk c, j < qlen} p[b][j] * hs[b][start+j][h]
// float4 per thread: 256 threads cover H=1024, fully coalesced 4KB row loads.
// Always writes its partial (zeros for empty chunks) -> deterministic, no init.
// ---------------------------------------------------------------------------
__global__ __launch_bounds__(256) void aa_k3_partial(
    const float* __restrict__ hs, const int* __restrict__ qi,
    const float* __restrict__ p, float* __restrict__ partial)
{
    const int c  = blockIdx.x;   // j-chunk
    const int b  = blockIdx.y;   // batch
    const int h0 = threadIdx.x * 4;

    const int start = qi[2 * b];
    const int qlen  = qi[2 * b + 1] - start;
    const int jbeg  = c * CHUNK_J;
    int jend = jbeg + CHUNK_J;
    jend = (jend < qlen) ? jend : qlen;

    const float*  base = hs + ((size_t)b * S_LEN + (size_t)start) * H_DIM + h0;
    const float*  pb   = p + (size_t)b * S_LEN;

    float4 acc = make_float4(0.f, 0.f, 0.f, 0.f);
    int j = jbeg;
    for (; j + 1 < jend; j += 2) {
        float pj0 = pb[j];
        float pj1 = pb[j + 1];
        float4 x0 = *(const float4*)(base + (size_t)j * H_DIM);
        float4 x1 = *(const float4*)(base + (size_t)(j + 1) * H_DIM);
        acc.x += pj0 * x0.x; acc.y += pj0 * x0.y;
        acc.z += pj0 * x0.z; acc.w += pj0 * x0.w;
        acc.x += pj1 * x1.x; acc.y += pj1 * x1.y;
        acc.z += pj1 * x1.z; acc.w += pj1 * x1.w;
    }
    if (j < jend) {
        float pj = pb[j];
        float4 x = *(const float4*)(base + (size_t)j * H_DIM);
        acc.x += pj * x.x; acc.y += pj * x.y;
        acc.z += pj * x.z; acc.w += pj * x.w;
    }

    float* op = partial + (((size_t)b * CHUNKS + c) * H_DIM) + h0;
    *(float4*)op = acc;
}

// ---------------------------------------------------------------------------
// K4: out[b][h] = sum_c partial[b][c][h]   (fixed order -> deterministic)
// ---------------------------------------------------------------------------
__global__ __launch_bounds__(256) void aa_k4_reduce(
    const float* __restrict__ partial, float* __restrict__ out)
{
    const int idx = blockIdx.x * 256 + threadIdx.x;   // 0 .. B*H-1
    const int b = idx >> 10;
    const int h = idx & (H_DIM - 1);
    float s = 0.0f;
    #pragma unroll
    for (int c = 0; c < CHUNKS; ++c)
        s += partial[((size_t)b * CHUNKS + c) * H_DIM + h];
    out[idx] = s;
}

extern "C" void kernel_launch(void* const* d_in, const int* in_sizes, int n_in,
                              void* d_out, int out_size, void* d_ws, size_t ws_size,
                              hipStream_t stream) {
    const float* hs = (const float*)d_in[0];   // (B, S, H) f32
    const int*   qi = (const int*)d_in[1];     // (B, 2) int
    const float* am = (const float*)d_in[2];   // (B,) f32
    const float* w  = (const float*)d_in[3];   // (H,) f32
    float* out = (float*)d_out;                // (B, H) f32

    // ws layout: [scores/probs: B*S f32 = 256KB][partials: B*CHUNKS*H f32 = 1MB]
    float* scores  = (float*)d_ws;
    float* partial = scores + (size_t)BATCH * S_LEN;

    // K1: WMMA f32 score GEMV. 16 rows/wave, 4 waves/block.
    aa_k1_scores<<<dim3(BATCH, S_LEN / 64), 128, 0, stream>>>(hs, qi, am, w, scores);
    // K2: per-batch softmax -> probabilities in place.
    aa_k2_softmax<<<BATCH, 256, 0, stream>>>(qi, scores);
    // K3: chunked weighted row-sum partials (float4 streaming).
    aa_k3_partial<<<dim3(CHUNKS, BATCH), 256, 0, stream>>>(hs, qi, scores, partial);
    // K4: reduce partials into the output.
    aa_k4_reduce<<<(BATCH * H_DIM) / 256, 256, 0, stream>>>(partial, out);
}